// Memory_38053410242909
// MI455X (gfx1250) — compile-verified
//
#include <hip/hip_runtime.h>
#include <math.h>

#define EPS 1e-8f
#define NN 8192
#define WW 128

typedef __attribute__((ext_vector_type(2))) float v2f;
typedef __attribute__((ext_vector_type(8))) float v8f;

static __device__ __forceinline__ float wave_reduce_add(float v) {
#pragma unroll
  for (int off = 16; off > 0; off >>= 1) v += __shfl_xor(v, off, 32);
  return v;
}

// ---------------------------------------------------------------------------
// K1: per-row ||memory_i||, memory_i . write_key  -> a[i] (softmax logit),
//     and read_vectors = memory^T @ read_weightings (atomic accumulate).
// One wave per row-group: each wave handles 32 rows, lane holds 4 columns.
// ---------------------------------------------------------------------------
__global__ __launch_bounds__(256) void k1_rownorm_rv(
    const float* __restrict__ mem, const float* __restrict__ wk,
    const float* __restrict__ wstr, const float* __restrict__ rw,
    float* __restrict__ a_ws, float* __restrict__ rv_out) {
  __shared__ float rvl[512];
  const int t = threadIdx.x;
  const int lane = t & 31, wave = t >> 5;
  for (int q = t; q < 512; q += 256) rvl[q] = 0.f;
  __syncthreads();

  const float4* wk4 = (const float4*)wk;
  float4 wkv = wk4[lane];
  float wksq = wave_reduce_add(wkv.x * wkv.x + wkv.y * wkv.y + wkv.z * wkv.z + wkv.w * wkv.w);
  float wknorm = sqrtf(wksq);
  float beta = wstr[0];

  const float4* mem4 = (const float4*)mem;
  const float4* rw4 = (const float4*)rw;

  float racc[4][4];
#pragma unroll
  for (int c = 0; c < 4; c++)
#pragma unroll
    for (int r = 0; r < 4; r++) racc[c][r] = 0.f;

  int base = blockIdx.x * 256 + wave * 32;
  for (int rr = 0; rr < 32; ++rr) {
    int i = base + rr;
    float4 m = mem4[(size_t)i * 32 + lane];
    float sq = m.x * m.x + m.y * m.y + m.z * m.z + m.w * m.w;
    float dt = m.x * wkv.x + m.y * wkv.y + m.z * wkv.z + m.w * wkv.w;
    sq = wave_reduce_add(sq);
    dt = wave_reduce_add(dt);
    if (lane == 0) a_ws[i] = dt / fmaxf(sqrtf(sq) * wknorm, EPS) * beta;
    float4 rwi = rw4[i];
    float mc[4] = {m.x, m.y, m.z, m.w};
    float rv[4] = {rwi.x, rwi.y, rwi.z, rwi.w};
#pragma unroll
    for (int c = 0; c < 4; c++)
#pragma unroll
      for (int r = 0; r < 4; r++) racc[c][r] = fmaf(mc[c], rv[r], racc[c][r]);
  }
#pragma unroll
  for (int c = 0; c < 4; c++)
#pragma unroll
    for (int r = 0; r < 4; r++) atomicAdd(&rvl[(lane * 4 + c) * 4 + r], racc[c][r]);
  __syncthreads();
  for (int q = t; q < 512; q += 256) atomicAdd(&rv_out[q], rvl[q]);
}

// ---------------------------------------------------------------------------
// K2: single-block small ops: bitonic argsort of usage (stable), cumprod scan,
// allocation weighting (gather alloc_sorted[order[k]] exactly as reference),
// softmax Z, write weighting ww, new_usage, new_prec, read-key norms.
// ---------------------------------------------------------------------------
__global__ __launch_bounds__(1024) void k2_small(
    const float* __restrict__ usage, const float* __restrict__ prec,
    const float* __restrict__ rw, const float* __restrict__ rkeys,
    const float* __restrict__ fgates, const float* __restrict__ agate,
    const float* __restrict__ wgate, const float* __restrict__ a_ws,
    float* __restrict__ ww_ws, float* __restrict__ knorm_ws,
    float* __restrict__ usage_out, float* __restrict__ prec_out) {
  __shared__ float su[NN];
  __shared__ unsigned short sidx[NN];
  __shared__ float sc[NN];
  __shared__ float red[32];
  const int t = threadIdx.x;

  for (int k = t; k < NN; k += 1024) {
    su[k] = usage[k];
    sidx[k] = (unsigned short)k;
  }
  __syncthreads();

  // bitonic sort ascending, stable via index tiebreak
  for (int size = 2; size <= NN; size <<= 1) {
    for (int stride = size >> 1; stride > 0; stride >>= 1) {
      for (int p = t; p < NN / 2; p += 1024) {
        int low = p & (stride - 1);
        int i = ((p - low) << 1) + low;
        int j = i + stride;
        float a = su[i], b = su[j];
        unsigned short ia = sidx[i], ib = sidx[j];
        bool up = ((i & size) == 0);
        bool agtb = (a > b) || (a == b && ia > ib);
        if (up ? agtb : !agtb) {
          su[i] = b; su[j] = a;
          sidx[i] = ib; sidx[j] = ia;
        }
      }
      __syncthreads();
    }
  }

  // inclusive cumprod scan of sorted usage into sc
  for (int k = t; k < NN; k += 1024) sc[k] = su[k];
  __syncthreads();
  for (int off = 1; off < NN; off <<= 1) {
    float tmp[8];
#pragma unroll
    for (int q = 0; q < 8; q++) {
      int k = t + q * 1024;
      tmp[q] = (k >= off) ? sc[k - off] : 1.f;
    }
    __syncthreads();
#pragma unroll
    for (int q = 0; q < 8; q++) {
      int k = t + q * 1024;
      sc[k] *= tmp[q];
    }
    __syncthreads();
  }
  // alloc_sorted into su: (1 - u_sorted[k]) * exclusive_prod[k]
  for (int k = t; k < NN; k += 1024) {
    float cps = (k == 0) ? 1.f : sc[k - 1];
    su[k] = (1.f - su[k]) * cps;
  }
  __syncthreads();

  // Z = sum(exp(a))  (a bounded in [-2,2], unshifted exp is safe)
  float zp = 0.f;
  for (int k = t; k < NN; k += 1024) zp += expf(a_ws[k]);
  zp = wave_reduce_add(zp);
  if ((t & 31) == 0) red[t >> 5] = zp;
  __syncthreads();
  {
    float v = (t < 32) ? red[t] : 0.f;
    v = wave_reduce_add(v);
    if (t == 0) red[0] = v;
  }
  __syncthreads();
  float Z = red[0];
  __syncthreads();

  float ga = agate[0], wg = wgate[0];
  float fg0 = fgates[0], fg1 = fgates[1], fg2 = fgates[2], fg3 = fgates[3];
  const float4* rw4 = (const float4*)rw;
  float invZ = 1.f / Z;
  float sp = 0.f;
  for (int k = t; k < NN; k += 1024) {
    float allocW = su[sidx[k]];  // reference: alloc_sorted[order[k]]
    float cw = expf(a_ws[k]) * invZ;
    float w = wg * (ga * allocW + (1.f - ga) * cw);
    ww_ws[k] = w;
    sp += w;
    float u = usage[k];
    float4 rk = rw4[k];
    float ret = (1.f - rk.x * fg0) * (1.f - rk.y * fg1) * (1.f - rk.z * fg2) * (1.f - rk.w * fg3);
    usage_out[k] = (u + w - u * w) * ret;
  }
  sp = wave_reduce_add(sp);
  if ((t & 31) == 0) red[t >> 5] = sp;
  __syncthreads();
  {
    float v = (t < 32) ? red[t] : 0.f;
    v = wave_reduce_add(v);
    if (t == 0) red[0] = v;
  }
  __syncthreads();
  float S = red[0];
  for (int k = t; k < NN; k += 1024) prec_out[k] = (1.f - S) * prec[k] + ww_ws[k];

  if (t < 4) {
    float s = 0.f;
    for (int w = 0; w < WW; w++) {
      float v = rkeys[w * 4 + t];
      s = fmaf(v, v, s);
    }
    knorm_ws[t] = sqrtf(s);
  }
}

// ---------------------------------------------------------------------------
// K3: new_memory = memory * (1 - ww (x) erase) + ww (x) write_vector
// ---------------------------------------------------------------------------
__global__ __launch_bounds__(256) void k3_memupdate(
    const float* __restrict__ mem, const float* __restrict__ erase,
    const float* __restrict__ wvec, const float* __restrict__ ww_ws,
    float* __restrict__ newmem) {
  int idx = blockIdx.x * 256 + threadIdx.x;  // float4 index, 0..262143
  int i = idx >> 5;
  int q = idx & 31;
  const float4* m4 = (const float4*)mem;
  const float4* e4 = (const float4*)erase;
  const float4* v4 = (const float4*)wvec;
  float4 m = m4[idx], e = e4[q], v = v4[q];
  float w = ww_ws[i];
  float4 o;
  o.x = fmaf(m.x, fmaf(-w, e.x, 1.f), w * v.x);
  o.y = fmaf(m.y, fmaf(-w, e.y, 1.f), w * v.y);
  o.z = fmaf(m.z, fmaf(-w, e.z, 1.f), w * v.z);
  o.w = fmaf(m.w, fmaf(-w, e.w, 1.f), w * v.w);
  ((float4*)newmem)[idx] = o;
}

// ---------------------------------------------------------------------------
// K4: THE bandwidth kernel. Fused over the 8192x8192 link matrix:
//   new_link = (1 - ww_i - ww_j)*link + ww_i*prec_j       (write 268MB)
//   bwd[i,r] += new_link[i,j]*rw[j,r]   (row sums, shuffle+LDS local)
//   fwd[j,r] += new_link[i,j]*rw[i,r]   (col sums, registers + atomics)
// Tile: 256 rows x 2048 cols, 256 threads, 8 cols/thread in registers.
// ---------------------------------------------------------------------------
__global__ __launch_bounds__(256) void k4_link(
    const float* __restrict__ link, const float* __restrict__ ww_ws,
    const float* __restrict__ prec, const float* __restrict__ rw,
    float* __restrict__ newlink, float* __restrict__ bwd_ws,
    float* __restrict__ fwd_ws) {
  __shared__ float wwj[2048];
  __shared__ float pj[2048];
  __shared__ float4 rwc[2048];
  __shared__ float wwi_s[256];
  __shared__ float4 rwr[256];
  __shared__ float sl[1024];
  const int t = threadIdx.x;
  const int lane = t & 31;
  const int j0 = blockIdx.x * 2048;
  const int i0 = blockIdx.y * 256;
  const float4* rw4 = (const float4*)rw;
  for (int k = t; k < 2048; k += 256) {
    wwj[k] = ww_ws[j0 + k];
    pj[k] = prec[j0 + k];
    rwc[k] = rw4[j0 + k];
  }
  wwi_s[t] = ww_ws[i0 + t];
  rwr[t] = rw4[i0 + t];
  for (int k = t; k < 1024; k += 256) sl[k] = 0.f;
  __syncthreads();

  float tacc[8][4];
#pragma unroll
  for (int c = 0; c < 8; c++)
#pragma unroll
    for (int r = 0; r < 4; r++) tacc[c][r] = 0.f;

  for (int ii = 0; ii < 256; ++ii) {
    const size_t rowoff = (size_t)(i0 + ii) * NN + j0;
    const float* Lr = link + rowoff;
    float* NLr = newlink + rowoff;
    if (ii < 248)  // rotating 8-row-lookahead prefetch -> global_prefetch_b8
      __builtin_prefetch(link + (size_t)(i0 + ii + 8) * NN + j0 + ((ii & 7) << 8) + t, 0, 0);
    float wwi = wwi_s[ii];
    float ci = 1.f - wwi;
    float4 ri = rwr[ii];
    float s0 = 0.f, s1 = 0.f, s2 = 0.f, s3 = 0.f;
#pragma unroll
    for (int c = 0; c < 8; c++) {
      int jl = t + (c << 8);
      float L = Lr[jl];
      float nl = fmaf(ci - wwj[jl], L, wwi * pj[jl]);
      NLr[jl] = nl;
      float4 rj = rwc[jl];
      s0 = fmaf(nl, rj.x, s0);
      s1 = fmaf(nl, rj.y, s1);
      s2 = fmaf(nl, rj.z, s2);
      s3 = fmaf(nl, rj.w, s3);
      tacc[c][0] = fmaf(nl, ri.x, tacc[c][0]);
      tacc[c][1] = fmaf(nl, ri.y, tacc[c][1]);
      tacc[c][2] = fmaf(nl, ri.z, tacc[c][2]);
      tacc[c][3] = fmaf(nl, ri.w, tacc[c][3]);
    }
    s0 = wave_reduce_add(s0);
    s1 = wave_reduce_add(s1);
    s2 = wave_reduce_add(s2);
    s3 = wave_reduce_add(s3);
    if (lane == 0) {
      atomicAdd(&sl[ii * 4 + 0], s0);
      atomicAdd(&sl[ii * 4 + 1], s1);
      atomicAdd(&sl[ii * 4 + 2], s2);
      atomicAdd(&sl[ii * 4 + 3], s3);
    }
  }
  __syncthreads();
  for (int q = t; q < 1024; q += 256) atomicAdd(&bwd_ws[i0 * 4 + q], sl[q]);
#pragma unroll
  for (int c = 0; c < 8; c++) {
    int j = j0 + t + (c << 8);
    atomicAdd(&fwd_ws[j * 4 + 0], tacc[c][0]);
    atomicAdd(&fwd_ws[j * 4 + 1], tacc[c][1]);
    atomicAdd(&fwd_ws[j * 4 + 2], tacc[c][2]);
    atomicAdd(&fwd_ws[j * 4 + 3], tacc[c][3]);
  }
}

// ---------------------------------------------------------------------------
// K5: sim = new_memory @ read_keys via v_wmma_f32_16x16x4_f32 (one wave per
// 16-row tile, K=128 -> 32 chained WMMAs), row norms via shfl, per-row
// softmax over R, then final new_rw combine with fwd/bwd and read_modes.
// B-fragment is built branchlessly (clamped LDS index * 0/1 mask) so EXEC
// stays all-1s through the WMMA loop with no save/restore sequences.
// ---------------------------------------------------------------------------
__global__ __launch_bounds__(256) void k5_sim_combine(
    const float* __restrict__ newmem, const float* __restrict__ rkeys,
    const float* __restrict__ rstr, const float* __restrict__ rmodes,
    const float* __restrict__ knorm_ws, const float* __restrict__ bwd_ws,
    const float* __restrict__ fwd_ws, float* __restrict__ newrw) {
  __shared__ float rkl[512];
  __shared__ float dotl[8][16][4];
  const int t = threadIdx.x;
  const int lane = t & 31, wave = t >> 5;
  for (int q = t; q < 512; q += 256) rkl[q] = rkeys[q];
  __syncthreads();

  const int r0 = blockIdx.x * 128 + wave * 16;
  const int mrow = lane & 15;
  const int khalf = (lane >> 4) << 1;  // lanes 0-15: K pair {0,1}; 16-31: {2,3}
  const float2* nm2 = (const float2*)(newmem + (size_t)(r0 + mrow) * WW);

  v8f acc = {};
  float nsq = 0.f;
  const float bmask = (mrow < 4) ? 1.f : 0.f;  // zero-pad B cols 4..15
  const int bcol = mrow & 3;                   // clamped in-range LDS column
  for (int k0 = 0; k0 < 128; k0 += 4) {
    float2 av = nm2[(k0 + khalf) >> 1];
    v2f A;
    A.x = av.x;
    A.y = av.y;
    // B (4x16): lane n holds read_keys[k][n] for n<4, zero elsewhere.
    v2f B;
    B.x = rkl[(k0 + khalf) * 4 + bcol] * bmask;
    B.y = rkl[(k0 + khalf + 1) * 4 + bcol] * bmask;
    acc = __builtin_amdgcn_wmma_f32_16x16x4_f32(false, A, false, B, (short)0, acc,
                                                false, false);
    nsq = fmaf(av.x, av.x, nsq);
    nsq = fmaf(av.y, av.y, nsq);
  }
  nsq += __shfl_xor(nsq, 16, 32);  // combine K-halves for same row

  // D layout: VGPR v, lanes 0-15 -> D[M=v][N=lane]; lanes 16-31 -> D[M=v+8]
  if (mrow < 4) {
    if (lane < 16) {
#pragma unroll
      for (int v = 0; v < 8; ++v) dotl[wave][v][mrow] = acc[v];
    } else {
#pragma unroll
      for (int v = 0; v < 8; ++v) dotl[wave][v + 8][mrow] = acc[v];
    }
  }
  __syncthreads();

  if (lane < 16) {
    int i = r0 + lane;
    float norm = sqrtf(nsq);
    float x[4], rc[4];
#pragma unroll
    for (int r = 0; r < 4; r++)
      x[r] = dotl[wave][lane][r] / (norm * knorm_ws[r]) * rstr[r];
    float mx = fmaxf(fmaxf(x[0], x[1]), fmaxf(x[2], x[3]));
    float s = 0.f;
#pragma unroll
    for (int r = 0; r < 4; r++) {
      rc[r] = expf(x[r] - mx);
      s += rc[r];
    }
    float inv = 1.f / s;
    const float4* b4 = (const float4*)bwd_ws;
    const float4* f4 = (const float4*)fwd_ws;
    float4 bw = b4[i], fw = f4[i];
    float4 o;
    o.x = bw.x * rmodes[0] + rc[0] * inv * rmodes[1] + fw.x * rmodes[2];
    o.y = bw.y * rmodes[3] + rc[1] * inv * rmodes[4] + fw.y * rmodes[5];
    o.z = bw.z * rmodes[6] + rc[2] * inv * rmodes[7] + fw.z * rmodes[8];
    o.w = bw.w * rmodes[9] + rc[3] * inv * rmodes[10] + fw.w * rmodes[11];
    ((float4*)newrw)[i] = o;
  }
}

// ---------------------------------------------------------------------------
extern "C" void kernel_launch(void* const* d_in, const int* in_sizes, int n_in,
                              void* d_out, int out_size, void* d_ws,
                              size_t ws_size, hipStream_t stream) {
  const float* mem = (const float*)d_in[0];
  const float* usage = (const float*)d_in[1];
  const float* prec = (const float*)d_in[2];
  const float* link = (const float*)d_in[3];
  const float* rw = (const float*)d_in[4];
  const float* rkeys = (const float*)d_in[5];
  const float* rstr = (const float*)d_in[6];
  const float* wkey = (const float*)d_in[7];
  const float* wstr = (const float*)d_in[8];
  const float* erase = (const float*)d_in[9];
  const float* wvec = (const float*)d_in[10];
  const float* fg = (const float*)d_in[11];
  const float* ag = (const float*)d_in[12];
  const float* wg = (const float*)d_in[13];
  const float* rmod = (const float*)d_in[14];

  float* out = (float*)d_out;
  float* rv_out = out;                          // W*R = 512
  float* newmem = out + 512;                    // N*W
  float* usage_out = newmem + (size_t)NN * WW;  // N
  float* prec_out = usage_out + NN;             // N
  float* newlink = prec_out + NN;               // N*N
  float* newrw = newlink + (size_t)NN * NN;     // N*R

  float* ws = (float*)d_ws;
  float* bwd_ws = ws;               // N*4
  float* fwd_ws = bwd_ws + NN * 4;  // N*4
  float* a_ws = fwd_ws + NN * 4;    // N
  float* ww_ws = a_ws + NN;         // N
  float* knorm = ww_ws + NN;        // 4

  hipMemsetAsync(rv_out, 0, 512 * sizeof(float), stream);
  hipMemsetAsync(bwd_ws, 0, (size_t)NN * 4 * sizeof(float), stream);
  hipMemsetAsync(fwd_ws, 0, (size_t)NN * 4 * sizeof(float), stream);

  k1_rownorm_rv<<<32, 256, 0, stream>>>(mem, wkey, wstr, rw, a_ws, rv_out);
  k2_small<<<1, 1024, 0, stream>>>(usage, prec, rw, rkeys, fg, ag, wg, a_ws,
                                   ww_ws, knorm, usage_out, prec_out);
  k3_memupdate<<<1024, 256, 0, stream>>>(mem, erase, wvec, ww_ws, newmem);
  k4_link<<<dim3(4, 32), 256, 0, stream>>>(link, ww_ws, prec, rw, newlink,
                                           bwd_ws, fwd_ws);
  k5_sim_combine<<<64, 256, 0, stream>>>(newmem, rkeys, rstr, rmod, knorm,
                                         bwd_ws, fwd_ws, newrw);
}